// SS3D_SPAIR_v1_19808389169237
// MI455X (gfx1250) — compile-verified
//
#include <hip/hip_runtime.h>
#include <stdint.h>

#define BLOCK 256
#define ITERS 4
#define TILE  (BLOCK * ITERS)

// CDNA5 async copy: global -> LDS, tracked with ASYNCcnt.
// vdst = LDS byte address (VGPR), vaddr = 64-bit global address (VGPR pair).
__device__ __forceinline__ void async_ld_b32(uint32_t lds_addr, const int* g) {
  asm volatile("global_load_async_to_lds_b32 %0, %1, off"
               :: "v"(lds_addr), "v"((unsigned long long)(uintptr_t)g)
               : "memory");
}

__device__ __forceinline__ void wait_async_le2() {
  asm volatile("s_wait_asynccnt 0x2" ::: "memory");
}
__device__ __forceinline__ void wait_async_le0() {
  asm volatile("s_wait_asynccnt 0x0" ::: "memory");
}

// 12-byte gather as a single global_load_b96 (dword-aligned wide load)
__device__ __forceinline__ void load_f3(float* v, const float* p) {
  __builtin_memcpy(v, p, 12);
}

__global__ __launch_bounds__(BLOCK)
void spair_fused(const float* __restrict__ pos,    // [N,3]
                 const float* __restrict__ gcen,   // [G,3]
                 const float* __restrict__ grat,   // [G,1]
                 const int*   __restrict__ batch,  // [N]
                 const int*   __restrict__ gidx,   // [E]
                 const int*   __restrict__ nidx,   // [E]
                 float* __restrict__ out,
                 int E) {
  __shared__ int sNi[2][BLOCK];
  __shared__ int sGi[2][BLOCK];
  const int tid  = (int)threadIdx.x;
  const int base = (int)blockIdx.x * TILE + tid;

  // LDS byte addresses for this lane's slots (low 32 bits of flat addr == LDS offset)
  const uint32_t aNi[2] = { (uint32_t)(uintptr_t)&sNi[0][tid],
                            (uint32_t)(uintptr_t)&sNi[1][tid] };
  const uint32_t aGi[2] = { (uint32_t)(uintptr_t)&sGi[0][tid],
                            (uint32_t)(uintptr_t)&sGi[1][tid] };

  // prologue: stage tile 0's index streams
  const int j0 = (base < E) ? base : 0;
  async_ld_b32(aNi[0], nidx + j0);
  async_ld_b32(aGi[0], gidx + j0);

  float* __restrict__ outL = out;                          // [E,3]
  float* __restrict__ outN = out + (size_t)3 * (size_t)E;  // norm*mask
  float* __restrict__ outW = out + (size_t)4 * (size_t)E;  // log boundary weight
  float* __restrict__ outB = out + (size_t)5 * (size_t)E;  // member_batch
  float* __restrict__ outM = out + (size_t)6 * (size_t)E;  // mask

#pragma unroll
  for (int k = 0; k < ITERS; ++k) {
    const int i   = base + k * BLOCK;
    const int buf = k & 1;

    if (k + 1 < ITERS) {
      int jn = i + BLOCK; jn = (jn < E) ? jn : 0;
      async_ld_b32(aNi[(k + 1) & 1], nidx + jn);
      async_ld_b32(aGi[(k + 1) & 1], gidx + jn);
      wait_async_le2();   // next tile's 2 ops may remain in flight; current tile done
    } else {
      wait_async_le0();
    }

    const int ni = sNi[buf][tid];
    const int gi = sGi[buf][tid];

    // gathers — one b96 each for pos/center, tables live in L2/WGP$
    float p[3], c[3];
    load_f3(p, pos  + 3 * (size_t)ni);
    load_f3(c, gcen + 3 * (size_t)gi);
    const float rr = grat[gi];
    const float bb = (float)batch[ni];

    const float r   = fmaf(0.15f, rr, 0.05f);   // (R_MAX-R_MIN)*ratio + R_MIN
    const float inv = 1.0f / r;
    const float dx = p[0] - c[0], dy = p[1] - c[1], dz = p[2] - c[2];
    const float lx = dx * inv, ly = dy * inv, lz = dz * inv;
    const float nrm = sqrtf(fmaf(lx, lx, fmaf(ly, ly, lz * lz)));
    const bool  m  = nrm < 1.2f;                 // 1 + B_RATIO
    const float mf = m ? 1.0f : 0.0f;
    // log(max(1 - (norm-1)/0.2, 0) + 1e-12), zeroed outside mask
    const float lw = m ? __logf(fmaxf(fmaf(-5.0f, nrm - 1.0f, 1.0f), 0.0f) + 1e-12f)
                       : 0.0f;
    const float s  = r * 5.0f * mf;              // local * r / R_MAX, masked

    if (i < E) {
      const size_t i3 = 3 * (size_t)i;
      __builtin_nontemporal_store(lx * s,        outL + i3 + 0);
      __builtin_nontemporal_store(ly * s,        outL + i3 + 1);
      __builtin_nontemporal_store(lz * s,        outL + i3 + 2);
      __builtin_nontemporal_store(nrm * mf,      outN + i);
      __builtin_nontemporal_store(lw,            outW + i);
      __builtin_nontemporal_store(m ? bb : -1.0f, outB + i);
      __builtin_nontemporal_store(mf,            outM + i);
    }
  }
}

extern "C" void kernel_launch(void* const* d_in, const int* in_sizes, int n_in,
                              void* d_out, int out_size, void* d_ws, size_t ws_size,
                              hipStream_t stream) {
  const float* pos  = (const float*)d_in[0];
  const float* gcen = (const float*)d_in[1];
  const float* grat = (const float*)d_in[2];
  const int*   bat  = (const int*)d_in[3];
  const int*   gidx = (const int*)d_in[4];
  const int*   nidx = (const int*)d_in[5];
  const int E = in_sizes[5];                 // 8,388,608 = 8192 * TILE exactly
  const int blocks = (E + TILE - 1) / TILE;
  spair_fused<<<blocks, BLOCK, 0, stream>>>(pos, gcen, grat, bat, gidx, nidx,
                                            (float*)d_out, E);
}